// SSC_KAN_SNN_15831249453631
// MI455X (gfx1250) — compile-verified
//
#include <hip/hip_runtime.h>

// ---------------------------------------------------------------------------
// SSC KAN-SNN forward, MI455X (gfx1250), wave32, f32 WMMA 16x16x4.
//
// Host loop over T=250 timesteps, 3 kernels per step (graph-capture safe).
// Layer 1 stages the KAN Bernstein basis panel in LDS (computed once per
// block instead of 8x redundantly per wave/half), then streams WMMA
// A-fragments from LDS (conflict-free ds_load_b64) against global B panels.
// ---------------------------------------------------------------------------

namespace {
constexpr int BATCH = 128, TSTEPS = 250, NIN = 700, H1 = 1024, H2 = 512, NOUT = 35;
constexpr int KKAN  = NIN * 4;          // 2800 (flattened (i,k) basis axis)
constexpr float THRESH = 1.0f;

// LDS chunking for the KAN basis panel: 4 chunks of 175 inputs (700 K each)
constexpr int ICHUNK = 175;             // inputs per chunk
constexpr int KCHUNK = ICHUNK * 4;      // 700 K values per chunk
constexpr int NCHUNK = NIN / ICHUNK;    // 4
constexpr int LDSPITCH = KCHUNK + 8;    // 708 floats: bank stride 4, conflict-free

// workspace offsets in floats (state only, ~2.9 MB)
constexpr size_t OFF_V1   = 0;
constexpr size_t OFF_A1   = OFF_V1   + (size_t)BATCH * H1;
constexpr size_t OFF_S1A  = OFF_A1   + (size_t)BATCH * H1;
constexpr size_t OFF_S1B  = OFF_S1A  + (size_t)BATCH * H1;
constexpr size_t OFF_V2   = OFF_S1B  + (size_t)BATCH * H1;
constexpr size_t OFF_A2   = OFF_V2   + (size_t)BATCH * H2;
constexpr size_t OFF_S2   = OFF_A2   + (size_t)BATCH * H2;
constexpr size_t OFF_VOUT = OFF_S2   + (size_t)BATCH * H2;
constexpr size_t OFF_OSUM = OFF_VOUT + (size_t)BATCH * NOUT;
constexpr size_t WS_FLOATS = OFF_OSUM + (size_t)BATCH * NOUT;    // 729,856
} // namespace

typedef __attribute__((ext_vector_type(2))) float v2f;
typedef __attribute__((ext_vector_type(8))) float v8f;

__device__ __forceinline__ v8f wmma_f32(v2f a, v2f b, v8f c) {
  // D(16x16) = A(16x4) * B(4x16) + C ; full-EXEC f32 matrix op
  return __builtin_amdgcn_wmma_f32_16x16x4_f32(false, a, false, b,
                                               (short)0, c, false, false);
}

// ---------------------------------------------------------------------------
// AdLIF elementwise update for one 16x16 D tile held in a v8f accumulator.
// Lane (lo,hi) owns column h, rows b0 + r + 8*hi (r = 0..7).
// ---------------------------------------------------------------------------
__device__ __forceinline__ void adlif_tile(
    v8f I, int b0, int h, int hi,
    const float* __restrict__ alpha, const float* __restrict__ rho,
    const float* __restrict__ beta,
    float* __restrict__ v, float* __restrict__ a,
    const float* __restrict__ sp, float* __restrict__ sn, int ld) {
  const float al = alpha[h], rh = rho[h], be = beta[h];
#pragma unroll
  for (int r = 0; r < 8; ++r) {
    const size_t idx = (size_t)(b0 + r + 8 * hi) * ld + h;
    const float vv = v[idx], aa = a[idx], spv = sp[idx];
    const float vn = al * (vv - spv * THRESH) + (1.0f - al) * (I[r] - aa);
    const float sk = (vn >= THRESH) ? 1.0f : 0.0f;
    v[idx]  = vn;
    a[idx]  = rh * aa + be * sk;
    sn[idx] = sk;
  }
}

// --------------------------- kernels ---------------------------------------

__global__ void zero_kernel(float* __restrict__ p, size_t n) {
  size_t i = (size_t)blockIdx.x * blockDim.x + threadIdx.x;
  if (i < n) p[i] = 0.0f;
}

// Layer 1: I1 = kan_basis(x_t)·C^T + spk1_prev·Wrec^T, then AdLIF.
// Wave = 16x32 output strip (2 accumulators). The basis panel for the block's
// 16 batch rows is staged chunk-wise in LDS by all 128 threads, then consumed
// as WMMA A-fragments via conflict-free ds_load_b64.
// grid (H1/128, BATCH/16), block 128 (4 waves)
__global__ __launch_bounds__(128)
void l1_kernel(const float* __restrict__ x, const float* __restrict__ C,
               const float* __restrict__ Wrec, const float* __restrict__ s1p,
               const float* __restrict__ alpha, const float* __restrict__ rho,
               const float* __restrict__ beta,
               float* __restrict__ v1, float* __restrict__ a1,
               float* __restrict__ s1n, int t) {
  __shared__ float sA[16 * LDSPITCH];   // 45.3 KB basis panel chunk

  const int lane = threadIdx.x & 31;
  const int wave = threadIdx.x >> 5;
  const int lo = lane & 15, hi = lane >> 4;
  const int bt = blockIdx.y;
  const int n0 = (blockIdx.x * 4 + wave) * 32;

  v8f acc0 = {}, acc1 = {};

  // ---- KAN GEMM over 4 K-chunks, basis staged in LDS ----
  const float* B0 = C + (size_t)(n0 + lo) * KKAN + 2 * hi;
  const float* B1 = C + (size_t)(n0 + 16 + lo) * KKAN + 2 * hi;
  for (int c = 0; c < NCHUNK; ++c) {
    // stage: 16 rows x 175 inputs -> 16 x 700 basis values
    for (int p = threadIdx.x; p < 16 * ICHUNK; p += 128) {
      const int row = p / ICHUNK;
      const int il  = p - row * ICHUNK;
      float xv = x[((size_t)(bt * 16 + row) * TSTEPS + t) * NIN + c * ICHUNK + il];
      xv = fminf(fmaxf(xv, 0.0f), 1.0f);
      const float om  = 1.0f - xv;
      const float om2 = om * om, x2 = xv * xv;
      float4 bs;
      bs.x = om2 * om;
      bs.y = 3.0f * xv * om2;
      bs.z = 3.0f * x2 * om;
      bs.w = x2 * xv;
      *(float4*)&sA[row * LDSPITCH + il * 4] = bs;
    }
    __syncthreads();

    const float* As  = &sA[lo * LDSPITCH + 2 * hi];
    const float* B0c = B0 + c * KCHUNK;
    const float* B1c = B1 + c * KCHUNK;
    for (int ko = 0; ko < KCHUNK; ko += 4) {
      v2f a = *(const v2f*)(As + ko);                 // ds_load_b64
      acc0 = wmma_f32(a, *(const v2f*)(B0c + ko), acc0);
      acc1 = wmma_f32(a, *(const v2f*)(B1c + ko), acc1);
    }
    __syncthreads();   // WAR fence before restaging
  }

  // ---- recurrent GEMM: spk1_prev · Wrec^T ----
  {
    const float* Sr = s1p + (size_t)(bt * 16 + lo) * H1 + 2 * hi;
    const float* R0 = Wrec + (size_t)(n0 + lo) * H1 + 2 * hi;
    const float* R1 = Wrec + (size_t)(n0 + 16 + lo) * H1 + 2 * hi;
    for (int ko = 0; ko < H1; ko += 4) {
      v2f a = *(const v2f*)(Sr + ko);
      acc0 = wmma_f32(a, *(const v2f*)(R0 + ko), acc0);
      acc1 = wmma_f32(a, *(const v2f*)(R1 + ko), acc1);
    }
  }
  adlif_tile(acc0, bt * 16, n0 + lo,      hi, alpha, rho, beta, v1, a1, s1p, s1n, H1);
  adlif_tile(acc1, bt * 16, n0 + 16 + lo, hi, alpha, rho, beta, v1, a1, s1p, s1n, H1);
}

// Layer 2: I2 = spk1·W2^T, AdLIF. grid (H2/128, BATCH/16), block 128
__global__ __launch_bounds__(128)
void l2_kernel(const float* __restrict__ s1, const float* __restrict__ W2,
               const float* __restrict__ alpha, const float* __restrict__ rho,
               const float* __restrict__ beta,
               float* __restrict__ v2, float* __restrict__ a2,
               float* __restrict__ s2) {
  const int lane = threadIdx.x & 31;
  const int wave = threadIdx.x >> 5;
  const int lo = lane & 15, hi = lane >> 4;
  const int bt = blockIdx.y;
  const int n0 = (blockIdx.x * 4 + wave) * 32;

  v8f acc0 = {}, acc1 = {};
  const float* Ar = s1 + (size_t)(bt * 16 + lo) * H1 + 2 * hi;
  const float* B0 = W2 + (size_t)(n0 + lo) * H1 + 2 * hi;
  const float* B1 = W2 + (size_t)(n0 + 16 + lo) * H1 + 2 * hi;
  for (int ko = 0; ko < H1; ko += 4) {
    v2f a = *(const v2f*)(Ar + ko);
    acc0 = wmma_f32(a, *(const v2f*)(B0 + ko), acc0);
    acc1 = wmma_f32(a, *(const v2f*)(B1 + ko), acc1);
  }
  adlif_tile(acc0, bt * 16, n0 + lo,      hi, alpha, rho, beta, v2, a2, s2, s2, H2);
  adlif_tile(acc1, bt * 16, n0 + 16 + lo, hi, alpha, rho, beta, v2, a2, s2, s2, H2);
}

// Layer 3: I3 = spk2·W3^T (N=35 padded to 48), leaky output integrator.
// grid (3, BATCH/16), block 32 (1 wave)
__global__ __launch_bounds__(32)
void l3_kernel(const float* __restrict__ s2, const float* __restrict__ W3,
               const float* __restrict__ bout,
               float* __restrict__ vout, float* __restrict__ osum) {
  const int lane = threadIdx.x & 31;
  const int lo = lane & 15, hi = lane >> 4;
  const int bt = blockIdx.y;
  const int n0 = blockIdx.x * 16;
  const int nrow = (n0 + lo < NOUT) ? (n0 + lo) : (NOUT - 1);  // clamp OOB rows

  v8f acc = {};
  const float* Ar = s2 + (size_t)(bt * 16 + lo) * H2 + 2 * hi;
  const float* Br = W3 + (size_t)nrow * H2 + 2 * hi;
  for (int ko = 0; ko < H2; ko += 4) {
    acc = wmma_f32(*(const v2f*)(Ar + ko), *(const v2f*)(Br + ko), acc);
  }
  const int o = n0 + lo;
  if (o < NOUT) {                 // divergence only after all WMMAs
    const float be = bout[o];
#pragma unroll
    for (int r = 0; r < 8; ++r) {
      const size_t idx = (size_t)(bt * 16 + r + 8 * hi) * NOUT + o;
      const float vo = be * vout[idx] + (1.0f - be) * acc[r];
      vout[idx] = vo;
      osum[idx] += vo;
    }
  }
}

__global__ void final_kernel(const float* __restrict__ osum,
                             float* __restrict__ out) {
  int i = blockIdx.x * 256 + threadIdx.x;
  if (i < BATCH * NOUT) out[i] = osum[i] * (1.0f / (float)TSTEPS);
}

// --------------------------- launch ----------------------------------------

extern "C" void kernel_launch(void* const* d_in, const int* in_sizes, int n_in,
                              void* d_out, int out_size, void* d_ws, size_t ws_size,
                              hipStream_t stream) {
  (void)in_sizes; (void)n_in; (void)out_size; (void)ws_size;
  const float* x       = (const float*)d_in[0];
  const float* C       = (const float*)d_in[1];
  const float* Wrec    = (const float*)d_in[2];
  const float* W2      = (const float*)d_in[3];
  const float* W3      = (const float*)d_in[4];
  const float* alpha1  = (const float*)d_in[5];
  const float* rho1    = (const float*)d_in[6];
  const float* beta_a1 = (const float*)d_in[7];
  const float* alpha2  = (const float*)d_in[8];
  const float* rho2    = (const float*)d_in[9];
  const float* beta_a2 = (const float*)d_in[10];
  const float* betao   = (const float*)d_in[11];

  float* ws   = (float*)d_ws;
  float* v1   = ws + OFF_V1;
  float* a1   = ws + OFF_A1;
  float* s1a  = ws + OFF_S1A;
  float* s1b  = ws + OFF_S1B;
  float* v2   = ws + OFF_V2;
  float* a2   = ws + OFF_A2;
  float* s2   = ws + OFF_S2;
  float* vout = ws + OFF_VOUT;
  float* osum = ws + OFF_OSUM;

  // zero all state (ws is poisoned by the harness)
  {
    int blocks = (int)((WS_FLOATS + 255) / 256);
    zero_kernel<<<blocks, 256, 0, stream>>>(ws, WS_FLOATS);
  }

  for (int t = 0; t < TSTEPS; ++t) {
    const float* s1_prev = (t & 1) ? s1b : s1a;   // double-buffered spikes
    float*       s1_next = (t & 1) ? s1a : s1b;

    l1_kernel<<<dim3(H1 / 128, BATCH / 16), 128, 0, stream>>>(
        x, C, Wrec, s1_prev, alpha1, rho1, beta_a1, v1, a1, s1_next, t);

    l2_kernel<<<dim3(H2 / 128, BATCH / 16), 128, 0, stream>>>(
        s1_next, W2, alpha2, rho2, beta_a2, v2, a2, s2);

    l3_kernel<<<dim3(3, BATCH / 16), 32, 0, stream>>>(s2, W3, betao, vout, osum);
  }

  final_kernel<<<(BATCH * NOUT + 255) / 256, 256, 0, stream>>>(osum, (float*)d_out);
}